// Comparator4Bit_23407571764118
// MI455X (gfx1250) — compile-verified
//
#include <hip/hip_runtime.h>
#include <cstdint>

#define BLOCK 256
#define ITERS 8
#define NBUF  4                         // 4-deep LDS ring -> 3 iterations of prefetch
#define ROWS_PER_BLOCK (BLOCK * ITERS)  // 2048 rows / block
#define BUF_BYTES (BLOCK * 32)          // 16B A-slot + 16B B-slot per thread

// low 32 bits of a flat LDS pointer == wave-relative LDS byte offset (aperture in [63:32])
__device__ __forceinline__ uint32_t lds_off(const void* p) {
  return (uint32_t)(uintptr_t)p;
}

// Inputs are exactly 0.0f (0x00000000) or 1.0f (0x3F800000): bit 23 is the boolean.
// Row layout is MSB-first: [a3,a2,a1,a0].
__device__ __forceinline__ void nibble_cmp(const uint4 av, const uint4 bv,
                                           float& gt, float& eq) {
  uint32_t ia = ((av.x >> 20) & 8u) | ((av.y >> 21) & 4u) |
                ((av.z >> 22) & 2u) | ((av.w >> 23) & 1u);
  uint32_t ib = ((bv.x >> 20) & 8u) | ((bv.y >> 21) & 4u) |
                ((bv.z >> 22) & 2u) | ((bv.w >> 23) & 1u);
  gt = (ia > ib)  ? 1.0f : 0.0f;
  eq = (ia == ib) ? 1.0f : 0.0f;
}

__global__ __launch_bounds__(BLOCK)
void cmp4_async_kernel(const float* __restrict__ A, const float* __restrict__ B,
                       float* __restrict__ outGt, float* __restrict__ outEq) {
  __shared__ __align__(16) char smem[NBUF * BUF_BYTES];

  const int tid = threadIdx.x;
  const long long rowBase = (long long)blockIdx.x * ROWS_PER_BLOCK;
  const char* aBase = (const char*)A + rowBase * 16;   // 16 B per row
  const char* bBase = (const char*)B + rowBase * 16;

  const uint32_t ldsA0 = lds_off(&smem[tid * 16]);
  const uint32_t ldsB0 = ldsA0 + BLOCK * 16;

  // Async copy one A-row and one B-row (16B each) into this lane's LDS slot.
  auto issue = [&](int it) {
    const uint32_t vofs = (uint32_t)(it * BLOCK + tid) * 16u;        // GVS: saddr + vgpr_i32
    const uint32_t buf  = (uint32_t)(it % NBUF) * BUF_BYTES;
    const uint32_t la = ldsA0 + buf;
    const uint32_t lb = ldsB0 + buf;
    asm volatile("global_load_async_to_lds_b128 %0, %1, %2"
                 :: "v"(la), "v"(vofs), "s"(aBase) : "memory");
    asm volatile("global_load_async_to_lds_b128 %0, %1, %2"
                 :: "v"(lb), "v"(vofs), "s"(bBase) : "memory");
  };

  // Prime the pipeline: NBUF-1 iterations in flight (6 async ops).
#pragma unroll
  for (int it = 0; it < NBUF - 1; ++it) issue(it);

#pragma unroll
  for (int it = 0; it < ITERS; ++it) {
    const bool more = (it + NBUF - 1) < ITERS;
    if (more) issue(it + NBUF - 1);
    // Async loads complete in order: waiting to <= 2*inflight guarantees
    // iteration `it`'s two copies have landed in LDS.
    const int inflight = more ? (NBUF - 1) : (ITERS - 1 - it);
    asm volatile("s_wait_asynccnt %0" :: "i"(2 * inflight) : "memory");

    const int buf = (it % NBUF) * BUF_BYTES;
    const uint4 av = *(const uint4*)&smem[buf + tid * 16];
    const uint4 bv = *(const uint4*)&smem[buf + BLOCK * 16 + tid * 16];

    float gt, eq;
    nibble_cmp(av, bv, gt, eq);

    const long long r = rowBase + it * BLOCK + tid;
    __builtin_nontemporal_store(gt, &outGt[r]);   // streamed once; keep out of L2
    __builtin_nontemporal_store(eq, &outEq[r]);
  }
}

// Generic remainder path (unused for N = 8388608, kept for correctness on any size).
__global__ void cmp4_tail_kernel(const float* __restrict__ A, const float* __restrict__ B,
                                 float* __restrict__ outGt, float* __restrict__ outEq,
                                 long long start, long long n) {
  long long r = start + (long long)blockIdx.x * blockDim.x + threadIdx.x;
  if (r >= n) return;
  const uint4 av = ((const uint4*)A)[r];
  const uint4 bv = ((const uint4*)B)[r];
  float gt, eq;
  nibble_cmp(av, bv, gt, eq);
  outGt[r] = gt;
  outEq[r] = eq;
}

extern "C" void kernel_launch(void* const* d_in, const int* in_sizes, int n_in,
                              void* d_out, int out_size, void* d_ws, size_t ws_size,
                              hipStream_t stream) {
  const float* A = (const float*)d_in[0];
  const float* B = (const float*)d_in[1];
  const long long n = (long long)in_sizes[0] / 4;   // rows
  float* outGt = (float*)d_out;
  float* outEq = outGt + n;

  const long long fullBlocks = n / ROWS_PER_BLOCK;  // 4096 for N = 8388608
  if (fullBlocks > 0) {
    cmp4_async_kernel<<<(unsigned)fullBlocks, BLOCK, 0, stream>>>(A, B, outGt, outEq);
  }
  const long long done = fullBlocks * ROWS_PER_BLOCK;
  const long long rem = n - done;
  if (rem > 0) {
    cmp4_tail_kernel<<<(unsigned)((rem + BLOCK - 1) / BLOCK), BLOCK, 0, stream>>>(
        A, B, outGt, outEq, done, n);
  }
}